// DistributionalDQN_15075335209627
// MI455X (gfx1250) — compile-verified
//
#include <hip/hip_runtime.h>
#include <hip/hip_bf16.h>

// ---------------------------------------------------------------------------
// C51 categorical projection for MI455X (gfx1250).
//
// Memory-bound kernel: ~218 MB of traffic -> ~9.4 us at 23.3 TB/s HBM.
// Strategy: thread-per-row with a private per-row 51-bin LDS histogram
// (ds_add_f32, no contention), prob tile staged global->LDS with the CDNA5
// async copy engine (ASYNCcnt), and fully coalesced non-temporal b128 output.
// ---------------------------------------------------------------------------

#define C51_ATOMS 51
#define C51_RPB   128                      // rows per block (4 waves of 32)
#define C51_VMIN  (-40.0f)
#define C51_VMAX  (80.0f)
#define C51_GAMMA (0.9f)
#define C51_DELTA (2.4f)                   // (VMAX - VMIN) / (ATOMS - 1)

typedef float __attribute__((ext_vector_type(4))) v4f;
typedef int   __attribute__((ext_vector_type(4))) v4i;

#if defined(__has_builtin)
#  if __has_builtin(__builtin_amdgcn_global_load_async_to_lds_b128)
#    define C51_HAVE_ASYNC_LDS 1
#  endif
#  if __has_builtin(__builtin_amdgcn_s_wait_asynccnt)
#    define C51_WAIT_ASYNC() __builtin_amdgcn_s_wait_asynccnt(0)
#  endif
#endif
#ifndef C51_HAVE_ASYNC_LDS
#  define C51_HAVE_ASYNC_LDS 0
#endif
#ifndef C51_WAIT_ASYNC
#  define C51_WAIT_ASYNC() asm volatile("s_wait_asynccnt 0" ::: "memory")
#endif

__global__ __launch_bounds__(C51_RPB) void c51_project_kernel(
    const float* __restrict__ prob_next,  // [B, 51]
    const float* __restrict__ r,          // [B]
    const float* __restrict__ done,       // [B]
    float* __restrict__ out)              // [B, 51]
{
    constexpr int TILE_ELEMS = C51_RPB * C51_ATOMS;   // 6528 floats (26112 B)
    constexpr int TILE_VEC4  = TILE_ELEMS / 4;        // 1632 (exact)

    // 26 KB staging + 26 KB accumulator = 52 KB -> ~6 blocks per 320 KB WGP.
    __shared__ __align__(16) float pst[TILE_ELEMS];
    __shared__ __align__(16) float acc[TILE_ELEMS];

    const int tid       = threadIdx.x;
    const int blockRow0 = blockIdx.x * C51_RPB;
    const int elemBase  = blockRow0 * C51_ATOMS;      // flat float index; *4B is 16B-aligned

    // ---- zero the accumulator tile (51 stores/thread, conflict-free) ----
    for (int i = tid; i < TILE_ELEMS; i += C51_RPB)
        acc[i] = 0.0f;

    // ---- stage prob_next tile into LDS, coalesced b128 ----
    const v4f* gp4 = reinterpret_cast<const v4f*>(prob_next + elemBase);
#if C51_HAVE_ASYNC_LDS
    {
        v4f* lp4 = reinterpret_cast<v4f*>(pst);
        for (int v = tid; v < TILE_VEC4; v += C51_RPB) {
            __builtin_amdgcn_global_load_async_to_lds_b128(
                (v4i*)(gp4 + v),             // global source (per-lane addr)
                (v4i*)(lp4 + v),             // LDS destination (per-lane addr)
                /*imm offset*/ 0, /*cpol*/ 0);
        }
        C51_WAIT_ASYNC();                    // my async copies done
    }
#else
    {
        v4f* lp4 = reinterpret_cast<v4f*>(pst);
        for (int v = tid; v < TILE_VEC4; v += C51_RPB)
            lp4[v] = __builtin_nontemporal_load(gp4 + v);
    }
#endif
    __syncthreads();                         // all waves' tiles visible

    // ---- per-row projection: scatter into private LDS histogram ----
    {
        const int   row = blockRow0 + tid;
        const float rr  = r[row];
        const float g   = C51_GAMMA * (1.0f - done[row]);
        const float* __restrict__ myP = pst + tid * C51_ATOMS; // stride 51 words:
        float*       __restrict__ myA = acc + tid * C51_ATOMS; // coprime w/ 64 banks

        #pragma unroll
        for (int j = 0; j < C51_ATOMS; ++j) {
            const float pj   = myP[j];
            const float sj   = C51_VMIN + C51_DELTA * (float)j;          // support
            float atom = fmaf(g, sj, rr);                                 // r + g*s_j
            atom = fminf(fmaxf(atom, C51_VMIN), C51_VMAX);                // clip
            const float b  = (atom - C51_VMIN) / C51_DELTA;               // in [0,50]
            const float lf = floorf(b);
            const float uf = ceilf(b);
            int li = (int)lf;
            int ui = (int)uf;
            li = li < 0 ? 0 : (li > C51_ATOMS - 1 ? C51_ATOMS - 1 : li);  // fp-edge guard
            ui = ui < 0 ? 0 : (ui > C51_ATOMS - 1 ? C51_ATOMS - 1 : ui);
            const float dml = (uf + ((lf == uf) ? 1.0f : 0.0f) - b) * pj;
            const float dmu = (b - lf) * pj;
            atomicAdd(&myA[li], dml);   // ds_add_f32 (no return) — private row,
            atomicAdd(&myA[ui], dmu);   // same-wave DS ops stay in order
        }
    }
    __syncthreads();

    // ---- coalesced non-temporal write-out (output never re-read) ----
    {
        const v4f* acc4 = reinterpret_cast<const v4f*>(acc);
        v4f*       out4 = reinterpret_cast<v4f*>(out + elemBase);
        for (int v = tid; v < TILE_VEC4; v += C51_RPB)
            __builtin_nontemporal_store(acc4[v], out4 + v);
    }
}

extern "C" void kernel_launch(void* const* d_in, const int* in_sizes, int n_in,
                              void* d_out, int out_size, void* d_ws, size_t ws_size,
                              hipStream_t stream) {
    const float* prob_next = (const float*)d_in[0];
    const float* r         = (const float*)d_in[1];
    const float* done      = (const float*)d_in[2];
    float*       out       = (float*)d_out;

    const int batch  = in_sizes[1];              // r has BATCH elements (524288)
    const int blocks = batch / C51_RPB;          // 524288 / 128 = 4096 (exact)

    c51_project_kernel<<<blocks, C51_RPB, 0, stream>>>(prob_next, r, done, out);
}